// MessagePassingLayer_14482629722854
// MI455X (gfx1250) — compile-verified
//
#include <hip/hip_runtime.h>

#define D_FEAT      128
#define WAVES_PB    8
#define TPB         (WAVES_PB * 32)
#define EDGES_PER_WAVE 8

// ---------------------------------------------------------------------------
// Kernel 1: zero the output accumulator (harness poisons d_out with 0xAA).
// Vectorized 128-bit stores; default RT hint keeps `out` resident in L2 for
// the atomic phase.
// ---------------------------------------------------------------------------
__global__ void mp_zero_kernel(float4* __restrict__ out, int n4) {
    int i = blockIdx.x * blockDim.x + threadIdx.x;
    if (i < n4) {
        out[i] = make_float4(0.f, 0.f, 0.f, 0.f);
    }
}

// ---------------------------------------------------------------------------
// Kernel 2: edge-parallel gather/scale/scatter-add.
//   One wave32 per edge (4 floats per lane = 512B row).
//   e rows are double-buffered through LDS via gfx1250 async-to-LDS loads
//   (ASYNCcnt pipe) with a non-temporal hint: e is a 307 MB one-shot stream,
//   so it must not evict the L2-resident hot set (r 51 MB + out 51 MB in the
//   192 MB L2). r gathers and out atomics stay RT / L2-resident.
// ---------------------------------------------------------------------------
__global__ void __launch_bounds__(TPB)
mp_edge_kernel(const float* __restrict__ r,
               const float* __restrict__ e,
               const int*   __restrict__ a,
               float*       __restrict__ out,
               int n_edges) {
    __shared__ __align__(16) float ebuf[WAVES_PB][2][D_FEAT];

    const int lane = threadIdx.x & 31;
    const int w    = threadIdx.x >> 5;

    // Force the edge id into an SGPR: wave-uniform control flow + one
    // coalesced index fetch per wave.
    int edge = __builtin_amdgcn_readfirstlane((int)(blockIdx.x * WAVES_PB + w));
    const int wstride = (int)(gridDim.x * WAVES_PB);

    if (edge >= n_edges) return;

    // Wave-private LDS byte offsets for the two staging buffers (lane slot =
    // 16 bytes). Low 32 bits of a generic pointer to LDS == DS-space address.
    const unsigned lds0 = (unsigned)(size_t)(&ebuf[w][0][lane * 4]);
    const unsigned lds1 = (unsigned)(size_t)(&ebuf[w][1][lane * 4]);

    // Prologue: kick off async stage of the first e row into buffer 0.
    {
        const float* gsrc = e + (size_t)edge * D_FEAT + lane * 4;
        asm volatile("global_load_async_to_lds_b128 %0, %1, off th:TH_LOAD_NT"
                     :: "v"(lds0), "v"(gsrc) : "memory");
    }

    int buf = 0;
    while (edge < n_edges) {
        const int next = edge + wstride;

        if (next < n_edges) {
            // Prefetch next edge's e row into the other buffer, then wait for
            // the *current* buffer only (asynccnt <= 1: newest still in flight).
            const float* gsrc = e + (size_t)next * D_FEAT + lane * 4;
            const unsigned nlds = buf ? lds0 : lds1;
            asm volatile("global_load_async_to_lds_b128 %0, %1, off th:TH_LOAD_NT"
                         :: "v"(nlds), "v"(gsrc) : "memory");
            asm volatile("s_wait_asynccnt 1" ::: "memory");
        } else {
            asm volatile("s_wait_asynccnt 0" ::: "memory");
        }

        // Edge endpoints: read once, non-temporal (a is a one-shot 10 MB
        // stream); identical address across the wave coalesces to one request.
        const int s = __builtin_nontemporal_load(a + 2 * edge);
        const int d = __builtin_nontemporal_load(a + 2 * edge + 1);

        // Staged edge features from LDS; r gathers hit the L2-resident copy.
        const float4 ev = *(const float4*)(&ebuf[w][buf][lane * 4]);
        const float4 rs = *(const float4*)(r + (size_t)s * D_FEAT + lane * 4);
        const float4 rd = *(const float4*)(r + (size_t)d * D_FEAT + lane * 4);

        // message r[s]*e scattered to node d; message r[d]*e scattered to s.
        float* od = out + (size_t)d * D_FEAT + lane * 4;
        float* os = out + (size_t)s * D_FEAT + lane * 4;

        unsafeAtomicAdd(od + 0, rs.x * ev.x);
        unsafeAtomicAdd(od + 1, rs.y * ev.y);
        unsafeAtomicAdd(od + 2, rs.z * ev.z);
        unsafeAtomicAdd(od + 3, rs.w * ev.w);

        unsafeAtomicAdd(os + 0, rd.x * ev.x);
        unsafeAtomicAdd(os + 1, rd.y * ev.y);
        unsafeAtomicAdd(os + 2, rd.z * ev.z);
        unsafeAtomicAdd(os + 3, rd.w * ev.w);

        edge = next;
        buf ^= 1;
    }
}

// ---------------------------------------------------------------------------
// Host entry point.
//   d_in[0] = r  [N_NODES * 128] f32
//   d_in[1] = e  [N_EDGES * 128] f32
//   d_in[2] = a  [N_EDGES * 2]   int32
//   d_out   = out [N_NODES * 128] f32
// ---------------------------------------------------------------------------
extern "C" void kernel_launch(void* const* d_in, const int* in_sizes, int n_in,
                              void* d_out, int out_size, void* d_ws, size_t ws_size,
                              hipStream_t stream) {
    const float* r = (const float*)d_in[0];
    const float* e = (const float*)d_in[1];
    const int*   a = (const int*)d_in[2];
    float* out = (float*)d_out;

    const int n_edges = in_sizes[2] / 2;

    // Zero the accumulator.
    const int n4 = out_size / 4;
    const int zblocks = (n4 + 255) / 256;
    mp_zero_kernel<<<zblocks, 256, 0, stream>>>((float4*)out, n4);

    // ~EDGES_PER_WAVE edges per wave so the async double-buffer has depth.
    const int total_waves = (n_edges + EDGES_PER_WAVE - 1) / EDGES_PER_WAVE;
    const int blocks = (total_waves + WAVES_PB - 1) / WAVES_PB;
    mp_edge_kernel<<<blocks, TPB, 0, stream>>>(r, e, a, out, n_edges);
}